// NoisyTopk_6889127542919
// MI455X (gfx1250) — compile-verified
//
#include <hip/hip_runtime.h>
#include <math.h>

typedef __attribute__((ext_vector_type(2))) float v2f;
typedef __attribute__((ext_vector_type(4))) float v4f;
typedef __attribute__((ext_vector_type(8))) float v8f;

#define D_DIM   2048
#define E_DIM   8
#define TOKENS  16384      // B*S = 4*4096
#define D_SLICE (D_DIM/2)  // each wave reduces half of D
#define TILES_PER_BLOCK 4  // 4 token tiles x 2 D-slices = 8 waves = 256 threads

// Two wave32s cooperate on one 16-token tile, each reducing half of D with
// V_WMMA_F32_16X16X4_F32:  A = combined [W;Wn] (16 x D), B = 16 token rows.
// C (16x16 f32): lane<16 -> logits[0..7] of token=lane (vgpr r = expert r)
//                lane>=16 -> noise[0..7]  of token=lane-16 (vgpr r = expert r)
__global__ __launch_bounds__(256) void noisy_topk_kernel(
    const float* __restrict__ mh,     // [TOKENS, D]
    const float* __restrict__ Wl,     // [8, D]
    const float* __restrict__ bl,     // [8]
    const float* __restrict__ Wn,     // [8, D]
    const float* __restrict__ bn,     // [8]
    const float* __restrict__ gauss,  // [TOKENS, 8]
    float* __restrict__ out_probs,    // [TOKENS, 8]
    float* __restrict__ out_idx)      // [TOKENS, 2] (as float values)
{
    __shared__ float red[TILES_PER_BLOCK][32][8];   // partial C from slice-1 waves

    const int tid   = threadIdx.x;
    const int lane  = tid & 31;
    const int w     = tid >> 5;        // wave in block: 0..7
    const int tile  = w >> 1;          // 0..3
    const int slice = w & 1;           // 0: k in [0,1024), 1: k in [1024,2048)

    const int tok_base = (blockIdx.x * TILES_PER_BLOCK + tile) * 16;

    const int half = lane >> 4;        // 0: lanes 0..15, 1: lanes 16..31
    const int l16  = lane & 15;
    const int koff = half * 2;         // K-pair per half-wave (ISA fragment layout)
    const int kbeg = slice * D_SLICE;

    // A operand: combined weight matrix, M = l16 (rows 0-7 = W, rows 8-15 = Wn)
    const float* arow = (l16 < 8) ? (Wl + (size_t)l16 * D_DIM)
                                  : (Wn + (size_t)(l16 - 8) * D_DIM);
    const float* aptr = arow + koff + kbeg;
    // B operand: token index is the N dimension
    const float* bptr = mh + (size_t)(tok_base + l16) * D_DIM + koff + kbeg;

    v8f acc = {};
    #pragma unroll 16
    for (int k = 0; k < D_SLICE; k += 4) {
        v2f a = *(const v2f*)(aptr + k);   // Wc[l16][kbeg+k+koff .. +1]
        v2f b = *(const v2f*)(bptr + k);   // mh[tok][kbeg+k+koff .. +1]
        acc = __builtin_amdgcn_wmma_f32_16x16x4_f32(
            /*neg_a=*/false, a, /*neg_b=*/false, b,
            /*c_mod=*/(short)0, acc, /*reuse_a=*/false, /*reuse_b=*/false);
    }

    // D-split reduction: slice-1 waves park partial C in LDS, slice-0 adds it.
    if (slice == 1) {
        v4f lo = { acc[0], acc[1], acc[2], acc[3] };
        v4f hi = { acc[4], acc[5], acc[6], acc[7] };
        *(v4f*)&red[tile][lane][0] = lo;
        *(v4f*)&red[tile][lane][4] = hi;
    }
    __syncthreads();
    if (slice == 1) return;

    {
        v4f lo = *(const v4f*)&red[tile][lane][0];
        v4f hi = *(const v4f*)&red[tile][lane][4];
        #pragma unroll
        for (int r = 0; r < 4; ++r) { acc[r] += lo[r]; acc[r + 4] += hi[r]; }
    }

    // Bias: logits half adds b[], noise half adds bn[] (tiny, cache resident)
    const float* biasp = half ? bn : bl;
    #pragma unroll
    for (int r = 0; r < E_DIM; ++r) acc[r] += biasp[r];

    // Move noise rows (lanes 16..31) next to logits (lanes 0..15)
    float logit[E_DIM], noise[E_DIM];
    #pragma unroll
    for (int r = 0; r < E_DIM; ++r) {
        logit[r] = acc[r];
        noise[r] = __shfl_xor(acc[r], 16, 32);
    }

    if (half == 0) {
        const int t = tok_base + l16;
        const v4f g0 = *(const v4f*)(gauss + (size_t)t * E_DIM);
        const v4f g1 = *(const v4f*)(gauss + (size_t)t * E_DIM + 4);

        // noisy = logit + gauss * softplus(noise)
        float nl[E_DIM];
        #pragma unroll
        for (int e = 0; e < E_DIM; ++e) {
            float x  = noise[e];
            float sp = (x > 20.0f) ? x : log1pf(__expf(x));
            float g  = (e < 4) ? g0[e] : g1[e - 4];
            nl[e] = logit[e] + g * sp;
        }

        // top-2 (first-occurrence wins on ties, like lax.top_k)
        float v1 = nl[0]; int i1 = 0;
        float v2 = -INFINITY; int i2 = 0;
        #pragma unroll
        for (int e = 1; e < E_DIM; ++e) {
            float v = nl[e];
            if (v > v1)      { v2 = v1; i2 = i1; v1 = v; i1 = e; }
            else if (v > v2) { v2 = v;  i2 = e; }
        }

        // softmax over the two kept values
        float e2  = __expf(v2 - v1);
        float inv = 1.0f / (1.0f + e2);
        float p1  = inv;
        float p2  = e2 * inv;

        // scatter back to dense E with zeros elsewhere
        v4f o0 = {}, o1 = {};
        #pragma unroll
        for (int e = 0; e < E_DIM; ++e) {
            float pv = (e == i1) ? p1 : ((e == i2) ? p2 : 0.0f);
            if (e < 4) o0[e] = pv; else o1[e - 4] = pv;
        }
        *(v4f*)(out_probs + (size_t)t * E_DIM)     = o0;
        *(v4f*)(out_probs + (size_t)t * E_DIM + 4) = o1;

        v2f oi = { (float)i1, (float)i2 };
        *(v2f*)(out_idx + (size_t)t * 2) = oi;
    }
}

extern "C" void kernel_launch(void* const* d_in, const int* in_sizes, int n_in,
                              void* d_out, int out_size, void* d_ws, size_t ws_size,
                              hipStream_t stream) {
    const float* mh    = (const float*)d_in[0];
    const float* Wl    = (const float*)d_in[1];
    const float* bl    = (const float*)d_in[2];
    const float* Wn    = (const float*)d_in[3];
    const float* bn    = (const float*)d_in[4];
    const float* gauss = (const float*)d_in[5];

    float* out       = (float*)d_out;
    float* out_probs = out;                          // [TOKENS,8]
    float* out_idx   = out + (size_t)TOKENS * E_DIM; // [TOKENS,2]

    // 8 waves/block = 4 token tiles x 2 D-slices -> 64 tokens/block
    dim3 block(256);
    dim3 grid(TOKENS / (TILES_PER_BLOCK * 16));      // 256 blocks, 2048 waves
    noisy_topk_kernel<<<grid, block, 0, stream>>>(mh, Wl, bl, Wn, bn, gauss,
                                                  out_probs, out_idx);
}